// FirstOrderAttention_89979564851538
// MI455X (gfx1250) — compile-verified
//
#include <hip/hip_runtime.h>
#include <hip/hip_bf16.h>

// MI455X (gfx1250) implementation.
// Roofline: gather traffic (1.18 GB) is L2-resident (value = 73.7 MB < 192 MB L2);
// HBM traffic ~0.4 GB -> ~17us floor; total ~26 GFLOP fp32 -> memory bound.
// Therefore keep fp32 and use V_WMMA_F32_16X16X4_F32 for every GEMM stage.
//
// Workspace layout (requires ~131.4 MB):
//   [0, 327680)                : 5 fused 128x128 f32 weight matrices (F0a,F0b,F1a,F1b,F2)
//   [327680, 327680+131072000) : a_buf [N*K, 128] f32 edge attention weights

typedef float v2f __attribute__((ext_vector_type(2)));
typedef float v8f __attribute__((ext_vector_type(8)));

static __device__ __forceinline__ v8f wmma4(v2f a, v2f b, v8f c) {
  // D = A(16x4 f32) * B(4x16 f32) + C(16x16 f32)
  return __builtin_amdgcn_wmma_f32_16x16x4_f32(
      /*neg_a=*/false, a, /*neg_b=*/false, b,
      /*c_mod=*/(short)0, c, /*reuse_a=*/false, /*reuse_b=*/false);
}

// ---------------------------------------------------------------------------
// Kernel 1: fuse TP weights with projection weights (5 x 128x128 GEMMs, tiny)
//   F0a = tp_w_same[0] @ proj_w[0]   F0b = tp_w_rdot @ proj_w[0]
//   F1a = tp_w_same[1] @ proj_w[1]   F1b = tp_w_rv0  @ proj_w[1]
//   F2  = tp_w_same[2] @ proj_w[2]
// ---------------------------------------------------------------------------
__global__ __launch_bounds__(256) void fuse_weights_kernel(
    const float* __restrict__ tp_w_same, const float* __restrict__ tp_w_rv0,
    const float* __restrict__ tp_w_rdot, const float* __restrict__ proj_w,
    float* __restrict__ F) {
  int gid = blockIdx.x * 256 + threadIdx.x;       // 5*16384 = 81920 total
  int mat = gid >> 14;
  int rc  = gid & 16383;
  int r = rc >> 7, d = rc & 127;
  const float* A;
  const float* B;
  if (mat == 0)      { A = tp_w_same;          B = proj_w;          }
  else if (mat == 1) { A = tp_w_rdot;          B = proj_w;          }
  else if (mat == 2) { A = tp_w_same + 16384;  B = proj_w + 16384;  }
  else if (mat == 3) { A = tp_w_rv0;           B = proj_w + 16384;  }
  else               { A = tp_w_same + 32768;  B = proj_w + 32768;  }
  float acc = 0.0f;
  for (int e = 0; e < 128; ++e) acc = fmaf(A[r * 128 + e], B[e * 128 + d], acc);
  F[gid] = acc;
}

// ---------------------------------------------------------------------------
// Kernel 2: per-edge radial MLP + head scaling -> a_buf[e][c]
//   64 edge-rows per block, 8 waves; wave w owns output columns [16w,16w+16).
//   GEMM1: [64x64]@[64x128] (WMMA f32 16x16x4, 16 k-steps)
//   LayerNorm + SiLU in LDS
//   GEMM2: [64x128]@[128x128] (32 k-steps)
//   epilogue: a = alpha[e,h] * inputhead / (edge_dis + 1e-8), h == w
// ---------------------------------------------------------------------------
__global__ __launch_bounds__(256) void edge_mlp_kernel(
    const float* __restrict__ x_edge, const float* __restrict__ w1,
    const float* __restrict__ b1, const float* __restrict__ ln_g,
    const float* __restrict__ ln_b, const float* __restrict__ w2,
    const float* __restrict__ b2, const float* __restrict__ alpha,
    const float* __restrict__ edge_dis, float* __restrict__ a_buf) {
  __shared__ float H1[64][130];  // padded stride: even (8B-aligned v2f) + bank spread

  const int tid  = threadIdx.x;
  const int lane = tid & 31;
  const int wv   = tid >> 5;     // wave id 0..7 == output column tile == head
  const int lm   = lane & 15;    // L % 16
  const int lh   = lane >> 4;    // L / 16
  const int col  = 16 * wv + lm;
  const long e0  = (long)blockIdx.x * 64;

  // ---- GEMM1: H1 = x_edge_tile @ w1 + b1 ----
  for (int mt = 0; mt < 4; ++mt) {
    v8f acc = {};
    const long e = e0 + mt * 16 + lm;            // A row (edge) for this lane
    for (int kt = 0; kt < 16; ++kt) {
      const int kk = 4 * kt + 2 * lh;
      v2f A = *(const v2f*)(x_edge + e * 64 + kk);
      v2f B;
      B.x = w1[kk * 128 + col];
      B.y = w1[(kk + 1) * 128 + col];
      acc = wmma4(A, B, acc);
    }
    const float bias = b1[col];
    for (int i = 0; i < 8; ++i)
      H1[mt * 16 + i + 8 * lh][col] = acc[i] + bias;   // D row = i + 8*(L/16)
  }
  __syncthreads();

  // ---- LayerNorm + SiLU (row-wise over 128) ----
  if (tid < 64) {
    const int row = tid;
    float mu = 0.0f;
    for (int c = 0; c < 128; ++c) mu += H1[row][c];
    mu *= (1.0f / 128.0f);
    float var = 0.0f;
    for (int c = 0; c < 128; ++c) { float d = H1[row][c] - mu; var = fmaf(d, d, var); }
    var *= (1.0f / 128.0f);
    const float inv = rsqrtf(var + 1e-5f);
    for (int c = 0; c < 128; ++c) {
      float xh = (H1[row][c] - mu) * inv * ln_g[c] + ln_b[c];
      H1[row][c] = xh / (1.0f + __expf(-xh));    // SiLU
    }
  }
  __syncthreads();

  // ---- GEMM2 + alpha/edge_dis scaling ----
  for (int mt = 0; mt < 4; ++mt) {
    v8f acc = {};
    for (int kt = 0; kt < 32; ++kt) {
      const int kk = 4 * kt + 2 * lh;
      v2f A = *(const v2f*)(&H1[mt * 16 + lm][kk]);
      v2f B;
      B.x = w2[kk * 128 + col];
      B.y = w2[(kk + 1) * 128 + col];
      acc = wmma4(A, B, acc);
    }
    const float bias = b2[col];
    for (int i = 0; i < 8; ++i) {
      const int row = mt * 16 + i + 8 * lh;
      const long e  = e0 + row;
      // a[n,k, h*16+j] = alpha[n,k,h] * inputhead / (dis+1e-8); h == wv here
      const float sc = alpha[e * 8 + wv] / (edge_dis[e] + 1e-8f);
      a_buf[e * 128 + col] = (acc[i] + bias) * sc;
    }
  }
}

// ---------------------------------------------------------------------------
// Kernel 3: gather V, reduce over k into 13 per-node feature rows, then apply
// fused per-degree weights with WMMA. 8 nodes per block (128 edges), 8 waves.
// Paths: 0..8 = aV[l], 9..11 = arV0[x], 12 = ardot.
// ---------------------------------------------------------------------------
__global__ __launch_bounds__(256) void gather_tp_proj_kernel(
    const float* __restrict__ a_buf, const float* __restrict__ value,
    const float* __restrict__ node_pos, const float* __restrict__ exp_node_pos,
    const int* __restrict__ outcell_index, const int* __restrict__ fidx,
    const float* __restrict__ F, const float* __restrict__ proj_b,
    float* __restrict__ out) {
  __shared__ float G[13][8][130];   // 54,080 B; even stride -> 8B-aligned v2f frags
  __shared__ float r_lds[128][3];
  __shared__ int   gidx_lds[128];

  const int tid = threadIdx.x;
  const int n0  = blockIdx.x * 8;
  const long e0 = (long)n0 * 16;

  // ---- setup: composed gather index + relative position per edge ----
  if (tid < 128) {
    const int f  = fidx[e0 + tid];
    gidx_lds[tid] = outcell_index[f];
    const int n = n0 + (tid >> 4);
    r_lds[tid][0] = node_pos[n * 3 + 0] - exp_node_pos[(long)f * 3 + 0];
    r_lds[tid][1] = node_pos[n * 3 + 1] - exp_node_pos[(long)f * 3 + 1];
    r_lds[tid][2] = node_pos[n * 3 + 2] - exp_node_pos[(long)f * 3 + 2];
  }
  __syncthreads();

  // ---- Phase R: gather-reduce (elementwise in c, sum over k) ----
  {
    const int c  = tid & 127;       // channel (contiguous per wave -> coalesced)
    const int hn = tid >> 7;        // node half (uniform per wave)
    for (int nn = 0; nn < 4; ++nn) {
      const int n = hn * 4 + nn;
      float acc[13];
      for (int p = 0; p < 13; ++p) acc[p] = 0.0f;
      for (int k = 0; k < 16; ++k) {
        const int el = n * 16 + k;
        const float a = a_buf[(e0 + el) * 128 + c];
        const float* vp = value + (long)gidx_lds[el] * 1152 + c;
        float v[9];
        for (int l = 0; l < 9; ++l) v[l] = vp[l * 128];
        const float rx = r_lds[el][0], ry = r_lds[el][1], rz = r_lds[el][2];
        for (int l = 0; l < 9; ++l) acc[l] = fmaf(a, v[l], acc[l]);
        const float rv0 = a * v[0];
        acc[9]  = fmaf(rx, rv0, acc[9]);
        acc[10] = fmaf(ry, rv0, acc[10]);
        acc[11] = fmaf(rz, rv0, acc[11]);
        acc[12] = fmaf(a, fmaf(rx, v[1], fmaf(ry, v[2], rz * v[3])), acc[12]);
      }
      for (int p = 0; p < 13; ++p) G[p][n][c] = acc[p];
    }
  }
  __syncthreads();

  // ---- Phase C: y_l = G_same(l) @ Fa(l) [+ G_extra(l) @ Fb(l)] [+ proj_b] ----
  const int lane = tid & 31;
  const int wv   = tid >> 5;
  const int lm   = lane & 15;
  const int lh   = lane >> 4;
  const int col  = 16 * wv + lm;
  const int mm   = lm & 7;          // clamp A rows 8..15 to dup of 0..7 (unused outputs)

  for (int l = 0; l < 9; ++l) {
    v8f acc = {};
    const float* Fa = (l == 0) ? F : (l < 4) ? (F + 2 * 16384) : (F + 4 * 16384);
    for (int kt = 0; kt < 32; ++kt) {
      const int kk = 4 * kt + 2 * lh;
      v2f A = *(const v2f*)(&G[l][mm][kk]);
      v2f B;
      B.x = Fa[kk * 128 + col];
      B.y = Fa[(kk + 1) * 128 + col];
      acc = wmma4(A, B, acc);
    }
    if (l < 4) {                    // uniform branch: EXEC stays full for WMMA
      const int p2 = (l == 0) ? 12 : (9 + l - 1);
      const float* Fb = (l == 0) ? (F + 16384) : (F + 3 * 16384);
      for (int kt = 0; kt < 32; ++kt) {
        const int kk = 4 * kt + 2 * lh;
        v2f A = *(const v2f*)(&G[p2][mm][kk]);
        v2f B;
        B.x = Fb[kk * 128 + col];
        B.y = Fb[(kk + 1) * 128 + col];
        acc = wmma4(A, B, acc);
      }
    }
    const float bias = (l == 0) ? proj_b[col] : 0.0f;
    if (lh == 0) {                  // lanes 0-15 hold D rows 0..7 (the 8 valid nodes)
      for (int i = 0; i < 8; ++i) {
        const long n = n0 + i;
        out[n * 1152 + l * 128 + col] = acc[i] + bias;
      }
    }
  }
}

// ---------------------------------------------------------------------------
extern "C" void kernel_launch(void* const* d_in, const int* in_sizes, int n_in,
                              void* d_out, int out_size, void* d_ws, size_t ws_size,
                              hipStream_t stream) {
  const float* alpha         = (const float*)d_in[0];
  const float* value         = (const float*)d_in[1];
  const float* x_edge        = (const float*)d_in[2];
  const float* node_pos      = (const float*)d_in[3];
  const float* edge_dis      = (const float*)d_in[4];
  const float* exp_node_pos  = (const float*)d_in[5];
  const float* rad_w1        = (const float*)d_in[6];
  const float* rad_b1        = (const float*)d_in[7];
  const float* ln_g          = (const float*)d_in[8];
  const float* ln_b          = (const float*)d_in[9];
  const float* rad_w2        = (const float*)d_in[10];
  const float* rad_b2        = (const float*)d_in[11];
  const float* tp_w_same     = (const float*)d_in[12];
  const float* tp_w_rv0      = (const float*)d_in[13];
  const float* tp_w_rdot     = (const float*)d_in[14];
  const float* proj_w        = (const float*)d_in[15];
  const float* proj_b        = (const float*)d_in[16];
  const int*   outcell_index = (const int*)d_in[17];
  const int*   fidx          = (const int*)d_in[18];
  float* out = (float*)d_out;

  const int n_edges = in_sizes[4];       // N*K = 256000
  const int N       = in_sizes[3] / 3;   // 16000

  float* F     = (float*)d_ws;                           // 5*16384 f32
  float* a_buf = (float*)((char*)d_ws + 5 * 16384 * 4);  // N*K*128 f32

  fuse_weights_kernel<<<(5 * 16384) / 256, 256, 0, stream>>>(
      tp_w_same, tp_w_rv0, tp_w_rdot, proj_w, F);

  edge_mlp_kernel<<<n_edges / 64, 256, 0, stream>>>(
      x_edge, rad_w1, rad_b1, ln_g, ln_b, rad_w2, rad_b2,
      alpha, edge_dis, a_buf);

  gather_tp_proj_kernel<<<N / 8, 256, 0, stream>>>(
      a_buf, value, node_pos, exp_node_pos, outcell_index, fidx,
      F, proj_b, out);
}